// AirNetworkPyTorch_12627203851115
// MI455X (gfx1250) — compile-verified
//
#include <hip/hip_runtime.h>
#include <math.h>

// ---------------------------------------------------------------------------
// AirNetwork solver for MI455X (gfx1250, wave32).
// 8192 independent fixed-point solves; batch-global done/stall semantics
// replicated exactly via a 3-pass scheme:
//   pass1: full 50-iter trajectories, per-(iter,block) max|res| (+ optional
//          per-iter flow snapshots when scratch allows)
//   pass2: one wave replays the scalar err/stall/done recurrence -> n_steps
//   pass3: final flow (snapshot load or re-simulation) -> 20 outputs/element
// ---------------------------------------------------------------------------

#define NB       4
#define N_OUTERI 50
#define N_INNERI 15
#define BLOCK    256

// dp(q) = A*q^2 + L*q   with q = |flow|
__device__ __forceinline__ void branch_dp4(const float A[NB], const float L[NB],
                                           const float f[NB], float p[NB]) {
#pragma unroll
  for (int j = 0; j < NB; ++j) {
    float aq = fabsf(f[j]);
    p[j] = fmaf(A[j], aq, L[j]) * aq;   // (A*aq + L)*aq
  }
}

// Reference _distribute: 15 iterations of proportional rebalancing.
// Rebalance uses the single-divide form:
//   x = target/(p/d + 1e-6) == target*d/(p + 1e-6*d),  d = f^2 + 1e-6
__device__ __forceinline__ void distribute(const float A[NB], const float L[NB],
                                           float total, float f[NB], float p[NB]) {
  float q0 = total * 0.25f;
#pragma unroll
  for (int j = 0; j < NB; ++j) f[j] = q0;
  for (int it = 0; it < N_INNERI; ++it) {
    float pv[NB], nv[NB];
    branch_dp4(A, L, f, pv);
    float tsum   = (pv[0] + pv[1]) + (pv[2] + pv[3]);
    float target = 0.25f * tsum;
    float s = 0.f;
#pragma unroll
    for (int j = 0; j < NB; ++j) {
      float d = fmaf(f[j], f[j], 1e-6f);
      float x = (target * d) / fmaf(1e-6f, d, pv[j]);
      nv[j] = sqrtf(fmaxf(x, 0.f));
      s += nv[j];
    }
    float sc = total / (s + 1e-6f);
#pragma unroll
    for (int j = 0; j < NB; ++j) f[j] = nv[j] * sc;
  }
  branch_dp4(A, L, f, p);
}

__device__ __forceinline__ float fan_pressure(float flow, float speed,
                                              float a1000, float b, float c) {
  float fn = flow / (speed + 1e-6f);
  float pd = fmaf(c, fn * fn, fmaf(b, fn, a1000));
  return fmaxf(pd * speed * speed, 0.f);
}

__device__ __forceinline__ float sys_pressure(float flow,
                                              const float sA[NB], const float sL[NB],
                                              const float eA[NB], const float eL[NB]) {
  float f[NB], p[NB];
  distribute(sA, sL, flow, f, p);
  float r = p[0];
  distribute(eA, eL, flow, f, p);
  return r + p[0];
}

// Per-element per-branch coefficients: A = exp(log_R_duct) + exp(dlo + dk*(1-pos))
__device__ __forceinline__ void load_side(const float* __restrict__ log_R_duct,
                                          const float* __restrict__ R_lin,
                                          const float* __restrict__ dlo,
                                          const float* __restrict__ dk,
                                          const float* __restrict__ pos,
                                          int b, float A[NB], float L[NB]) {
#pragma unroll
  for (int j = 0; j < NB; ++j) {
    float rduct = expf(log_R_duct[j]);                    // uniform -> s_load path
    float pj    = pos[b * NB + j];
    float rdamp = expf(fmaf(dk[j], 1.f - pj, dlo[j]));
    A[j] = rduct + rdamp;
    L[j] = fabsf(R_lin[j]);
  }
}

__device__ __forceinline__ float outer_alpha(int t) {
  return fmaxf(0.5f * powf(0.95f, (float)t), 0.05f);
}

// ---------------------------------------------------------------------------
// Pass 1
// ---------------------------------------------------------------------------
__global__ void __launch_bounds__(BLOCK)
air_pass1(const float* __restrict__ fan_speed,
          const float* __restrict__ spos, const float* __restrict__ epos,
          const float* __restrict__ fa, const float* __restrict__ fb,
          const float* __restrict__ fc,
          const float* __restrict__ slr, const float* __restrict__ srl,
          const float* __restrict__ sdo, const float* __restrict__ sdk,
          const float* __restrict__ elr, const float* __restrict__ erl,
          const float* __restrict__ edo, const float* __restrict__ edk,
          float* __restrict__ blockmax, float* __restrict__ snaps, int B) {
  __shared__ float red[BLOCK / 32];
  int b  = blockIdx.x * blockDim.x + threadIdx.x;
  int bb = b < B ? b : B - 1;                      // duplicate tail: harmless for max

  __builtin_prefetch(fan_speed + bb, 0, 1);        // global_prefetch_b8
  __builtin_prefetch(spos + bb * NB, 0, 1);
  __builtin_prefetch(epos + bb * NB, 0, 1);

  float sA[NB], sL[NB], eA[NB], eL[NB];
  load_side(slr, srl, sdo, sdk, spos, bb, sA, sL);
  load_side(elr, erl, edo, edk, epos, bb, eA, eL);
  const float speed = fan_speed[bb];
  const float a1000 = fa[0] * 1000.f, bv = fb[0], cv = fc[0];
  const float flow_scale = 2.0f / (500.0f + 1e-6f);

  float flow = speed * 2.0f;
  const int lane = threadIdx.x & 31;
  const int wv   = threadIdx.x >> 5;

  for (int t = 0; t < N_OUTERI; ++t) {
    float res  = fan_pressure(flow, speed, a1000, bv, cv)
               - sys_pressure(flow, sA, sL, eA, eL);
    float m = fabsf(res);
#pragma unroll
    for (int k = 16; k >= 1; k >>= 1) m = fmaxf(m, __shfl_xor(m, k, 32));
    if (lane == 0) red[wv] = m;
    __syncthreads();                                // s_barrier_signal/-wait
    if (wv == 0) {
      float v = (lane < (BLOCK / 32)) ? red[lane] : 0.f;
#pragma unroll
      for (int k = 4; k >= 1; k >>= 1) v = fmaxf(v, __shfl_xor(v, k, 32));
      if (lane == 0) blockmax[t * gridDim.x + blockIdx.x] = v;
    }
    __syncthreads();
    flow = fminf(fmaxf(fmaf(outer_alpha(t) * flow_scale, res, flow), 0.01f), 3.0f);
    if (snaps && b < B) snaps[t * B + b] = flow;    // coalesced trajectory snapshot
  }
}

// ---------------------------------------------------------------------------
// Pass 2: one wave. Reduce block maxima -> err_t, replay scalar stall/done
//         recurrence, emit n_steps (= #updates actually applied).
// ---------------------------------------------------------------------------
__global__ void air_pass2(const float* __restrict__ blockmax, int nblocks,
                          int* __restrict__ n_steps) {
  const int lane = threadIdx.x;
  float prev_err = INFINITY;
  int   stall = 0;
  int   steps = N_OUTERI;
  bool  done  = false;
  for (int t = 0; t < N_OUTERI; ++t) {
    float m = -INFINITY;
    for (int j = lane; j < nblocks; j += 32) m = fmaxf(m, blockmax[t * nblocks + j]);
#pragma unroll
    for (int k = 16; k >= 1; k >>= 1) m = fmaxf(m, __shfl_xor(m, k, 32));
    float err = m;                                   // uniform across lanes now
    bool stalled = fabsf(err - prev_err) < 1e-6f;
    stall = stalled ? stall + 1 : 0;
    bool nd = done || (err < 1e-3f) || (stall > 10);
    if (nd && !done) steps = t;                      // update at t is skipped
    done = nd;
    if (!done) prev_err = err;
    if (done) break;                                 // sticky: flows frozen forever
  }
  if (lane == 0) *n_steps = steps;
}

// ---------------------------------------------------------------------------
// Output epilogue shared by both pass-3 variants.
// ---------------------------------------------------------------------------
__device__ __forceinline__ void write_outputs(int b, float flow, float speed,
                                              float a1000, float bv, float cv,
                                              const float sA[NB], const float sL[NB],
                                              const float eA[NB], const float eL[NB],
                                              float* __restrict__ out) {
  flow = fmaxf(flow, 0.f);
  float fan_p = fan_pressure(flow, speed, a1000, bv, cv);
  float sf[NB], sp[NB], ef[NB], ep[NB];
  distribute(sA, sL, flow, sf, sp);
  distribute(eA, eL, flow, ef, ep);
  float sys_p = sp[0] + ep[0];

  float fr = flow / (2.0f + 1e-6f);
  float pr = fan_p / (500.0f + 1e-6f);
  float eta = 0.65f * (1.f - 0.3f * (fr - 1.f) * (fr - 1.f)
                           - 0.2f * (pr - 1.f) * (pr - 1.f));
  eta = fminf(fmaxf(eta, 0.3f), 0.75f);
  float power = flow * fan_p / (eta * 0.9f + 1e-6f);

  float4* dst = (float4*)(out + (size_t)b * 20);     // b*80B is 16B aligned
  dst[0] = make_float4(flow, fan_p, sys_p, power);
  dst[1] = make_float4(sf[0], sf[1], sf[2], sf[3]);
  dst[2] = make_float4(sp[0], sp[1], sp[2], sp[3]);
  dst[3] = make_float4(ef[0], ef[1], ef[2], ef[3]);
  dst[4] = make_float4(ep[0], ep[1], ep[2], ep[3]);
}

// ---------------------------------------------------------------------------
// Pass 3a: snapshot path — final flow is a single coalesced load.
// ---------------------------------------------------------------------------
__global__ void __launch_bounds__(BLOCK)
air_pass3_snap(const float* __restrict__ fan_speed,
               const float* __restrict__ spos, const float* __restrict__ epos,
               const float* __restrict__ fa, const float* __restrict__ fb,
               const float* __restrict__ fc,
               const float* __restrict__ slr, const float* __restrict__ srl,
               const float* __restrict__ sdo, const float* __restrict__ sdk,
               const float* __restrict__ elr, const float* __restrict__ erl,
               const float* __restrict__ edo, const float* __restrict__ edk,
               const int* __restrict__ n_steps_p, const float* __restrict__ snaps,
               float* __restrict__ out, int B) {
  int b = blockIdx.x * blockDim.x + threadIdx.x;
  if (b >= B) return;
  float sA[NB], sL[NB], eA[NB], eL[NB];
  load_side(slr, srl, sdo, sdk, spos, b, sA, sL);
  load_side(elr, erl, edo, edk, epos, b, eA, eL);
  const float speed = fan_speed[b];
  const float a1000 = fa[0] * 1000.f, bv = fb[0], cv = fc[0];
  const int nsteps = *n_steps_p;                     // uniform
  float flow = (nsteps == 0) ? speed * 2.0f : snaps[(size_t)(nsteps - 1) * B + b];
  write_outputs(b, flow, speed, a1000, bv, cv, sA, sL, eA, eL, out);
}

// ---------------------------------------------------------------------------
// Pass 3b: fallback — re-simulate n_steps updates (uniform trip count).
// ---------------------------------------------------------------------------
__global__ void __launch_bounds__(BLOCK)
air_pass3_resim(const float* __restrict__ fan_speed,
                const float* __restrict__ spos, const float* __restrict__ epos,
                const float* __restrict__ fa, const float* __restrict__ fb,
                const float* __restrict__ fc,
                const float* __restrict__ slr, const float* __restrict__ srl,
                const float* __restrict__ sdo, const float* __restrict__ sdk,
                const float* __restrict__ elr, const float* __restrict__ erl,
                const float* __restrict__ edo, const float* __restrict__ edk,
                const int* __restrict__ n_steps_p, float* __restrict__ out, int B) {
  int b = blockIdx.x * blockDim.x + threadIdx.x;
  if (b >= B) return;
  float sA[NB], sL[NB], eA[NB], eL[NB];
  load_side(slr, srl, sdo, sdk, spos, b, sA, sL);
  load_side(elr, erl, edo, edk, epos, b, eA, eL);
  const float speed = fan_speed[b];
  const float a1000 = fa[0] * 1000.f, bv = fb[0], cv = fc[0];
  const float flow_scale = 2.0f / (500.0f + 1e-6f);
  const int nsteps = *n_steps_p;                     // uniform across grid

  float flow = speed * 2.0f;
  for (int t = 0; t < nsteps; ++t) {
    float res = fan_pressure(flow, speed, a1000, bv, cv)
              - sys_pressure(flow, sA, sL, eA, eL);
    flow = fminf(fmaxf(fmaf(outer_alpha(t) * flow_scale, res, flow), 0.01f), 3.0f);
  }
  write_outputs(b, flow, speed, a1000, bv, cv, sA, sL, eA, eL, out);
}

// ---------------------------------------------------------------------------
extern "C" void kernel_launch(void* const* d_in, const int* in_sizes, int n_in,
                              void* d_out, int out_size, void* d_ws, size_t ws_size,
                              hipStream_t stream) {
  const float* fan_speed = (const float*)d_in[0];
  const float* spos      = (const float*)d_in[1];
  const float* epos      = (const float*)d_in[2];
  const float* fa        = (const float*)d_in[3];
  const float* fb        = (const float*)d_in[4];
  const float* fc        = (const float*)d_in[5];
  const float* slr       = (const float*)d_in[6];
  const float* srl       = (const float*)d_in[7];
  const float* sdo       = (const float*)d_in[8];
  const float* sdk       = (const float*)d_in[9];
  const float* elr       = (const float*)d_in[10];
  const float* erl       = (const float*)d_in[11];
  const float* edo       = (const float*)d_in[12];
  const float* edk       = (const float*)d_in[13];
  float*       out       = (float*)d_out;

  const int B       = in_sizes[0];
  const int nblocks = (B + BLOCK - 1) / BLOCK;

  // Workspace layout: [blockmax: 50*nblocks f32][n_steps: i32][pad][snaps: 50*B f32]
  size_t off_nsteps = sizeof(float) * (size_t)N_OUTERI * nblocks;
  size_t off_snaps  = (off_nsteps + sizeof(int) + 15) & ~(size_t)15;
  size_t need_snap  = off_snaps + sizeof(float) * (size_t)N_OUTERI * B;

  float* blockmax = (float*)d_ws;
  int*   n_steps  = (int*)((char*)d_ws + off_nsteps);
  bool   use_snap = (ws_size >= need_snap);          // fixed per harness -> deterministic
  float* snaps    = use_snap ? (float*)((char*)d_ws + off_snaps) : nullptr;

  air_pass1<<<nblocks, BLOCK, 0, stream>>>(fan_speed, spos, epos, fa, fb, fc,
                                           slr, srl, sdo, sdk, elr, erl, edo, edk,
                                           blockmax, snaps, B);
  air_pass2<<<1, 32, 0, stream>>>(blockmax, nblocks, n_steps);
  if (use_snap) {
    air_pass3_snap<<<nblocks, BLOCK, 0, stream>>>(fan_speed, spos, epos, fa, fb, fc,
                                                  slr, srl, sdo, sdk, elr, erl, edo, edk,
                                                  n_steps, snaps, out, B);
  } else {
    air_pass3_resim<<<nblocks, BLOCK, 0, stream>>>(fan_speed, spos, epos, fa, fb, fc,
                                                   slr, srl, sdo, sdk, elr, erl, edo, edk,
                                                   n_steps, out, B);
  }
  (void)n_in; (void)out_size;
}